// DGCNN_446676599447
// MI455X (gfx1250) — compile-verified
//
#include <hip/hip_runtime.h>
#include <hip/hip_bf16.h>

#define KNN   20
#define NPTS  2048
#define BATCH 8
#define KC    32   // K-chunk (floats) for the pipelined GEMM path

typedef float v2f __attribute__((ext_vector_type(2)));
typedef float v8f __attribute__((ext_vector_type(8)));

// D = A(16x4 f32) * B(4x16 f32) + C(16x16 f32), wave32 WMMA.
__device__ __forceinline__ v8f wmma_f32_16x16x4(v2f a, v2f b, v8f c) {
  return __builtin_amdgcn_wmma_f32_16x16x4_f32(false, a, false, b, (short)0, c,
                                               false, false);
}

__device__ __forceinline__ float lrelu(float x) { return x > 0.f ? x : 0.2f * x; }

// Monotone float -> uint encoding so float max == uint max. enc(any real) > 0.
__device__ __forceinline__ unsigned fenc(float f) {
  unsigned u = __float_as_uint(f);
  return (u & 0x80000000u) ? ~u : (u | 0x80000000u);
}
__device__ __forceinline__ float fdec(unsigned e) {
  return (e & 0x80000000u) ? __uint_as_float(e & 0x7fffffffu)
                           : __uint_as_float(~e);
}

// ---- transpose x (B,3,N) -> xt4 (B*N, 4) with zero pad channel -------------
__global__ void k_build_xt4(const float* __restrict__ x, float* __restrict__ xt4) {
  int i = blockIdx.x * blockDim.x + threadIdx.x;   // 0 .. B*N-1
  int b = i / NPTS, n = i % NPTS;
  const float* xb = x + (size_t)b * 3 * NPTS;
  float4 v;
  v.x = xb[0 * NPTS + n];
  v.y = xb[1 * NPTS + n];
  v.z = xb[2 * NPTS + n];
  v.w = 0.f;
  *(float4*)(xt4 + (size_t)i * 4) = v;
}

// ---- row squared norms -----------------------------------------------------
__global__ void k_xx(const float* __restrict__ X, int ldx, int C,
                     float* __restrict__ xx) {
  int i = blockIdx.x * blockDim.x + threadIdx.x;   // 0 .. B*N-1
  const float* r = X + (size_t)i * ldx;
  float s = 0.f;
  for (int c = 0; c < C; ++c) { float v = r[c]; s += v * v; }
  xx[i] = s;
}

// ---- fused pairwise-distance (WMMA) + top-20 selection ---------------------
// grid: (N/16, B), block: 256 (8 waves). Each workgroup owns 16 rows.
// A-tile register-resident; per round all B loads batched ahead of the WMMA
// chain. Top-20 lists live in LDS; only the cached 20th-best value (vmin)
// stays in a register, so the candidate scan is one VALU compare.
template <int C>
__global__ __launch_bounds__(256, 1) void k_knn_tpl(
    const float* __restrict__ X, int ldx,
    const float* __restrict__ xx, int* __restrict__ idxo) {
  __shared__ float cand[8][16][16];
  __shared__ float svals[16][KNN];
  __shared__ int   sinds[16][KNN];
  int tid = threadIdx.x;
  int wave = tid >> 5, lane = tid & 31;
  int half = lane >> 4, l15 = lane & 15;
  int b = blockIdx.y;
  int r0 = blockIdx.x * 16;
  const float* Xb  = X  + (size_t)b * NPTS * ldx;
  const float* xxb = xx + (size_t)b * NPTS;
  const float* arow = Xb + (size_t)(r0 + l15) * ldx + half * 2;

  for (int i = tid; i < 16 * KNN; i += 256) {
    ((float*)svals)[i] = -3.4e38f;
    ((int*)sinds)[i] = 0;
  }

  v2f areg[C / 4];
#pragma unroll
  for (int s = 0; s < C / 4; ++s) areg[s] = *(const v2f*)(arow + 4 * s);

  float rn[8];
#pragma unroll
  for (int r = 0; r < 8; ++r) rn[r] = xxb[r0 + r + 8 * half];

  bool sel = ((tid & 15) == 0);
  int selrow = tid >> 4;   // 0..15
  float vmin = -3.4e38f;
  __syncthreads();

  for (int t = 0; t < NPTS / (16 * 8); ++t) {   // 16 rounds x 8 waves x 16 cols
    int j0 = (t * 8 + wave) * 16;
    const float* brow = Xb + (size_t)(j0 + l15) * ldx + half * 2;
    v2f breg[C / 4];
#pragma unroll
    for (int s = 0; s < C / 4; ++s) breg[s] = *(const v2f*)(brow + 4 * s);
    v8f acc = {};
#pragma unroll
    for (int s = 0; s < C / 4; ++s) acc = wmma_f32_16x16x4(areg[s], breg[s], acc);

    float xxc = xxb[j0 + l15];               // this lane's column norm
#pragma unroll
    for (int r = 0; r < 8; ++r)              // pd = -|xi|^2 + 2 xi.xj - |xj|^2
      cand[wave][r + 8 * half][l15] = 2.f * acc[r] - rn[r] - xxc;
    __syncthreads();

    if (sel) {
      for (int w = 0; w < 8; ++w) {
        int jb = (t * 8 + w) * 16;
#pragma unroll
        for (int c = 0; c < 16; ++c) {
          float v = cand[w][selrow][c];
          if (v > vmin) {                    // rare: insert into LDS top-20
            svals[selrow][KNN - 1] = v;
            sinds[selrow][KNN - 1] = jb + c;
#pragma unroll
            for (int q = KNN - 1; q > 0; --q) {
              float lo = svals[selrow][q], hi = svals[selrow][q - 1];
              if (lo > hi) {
                svals[selrow][q] = hi; svals[selrow][q - 1] = lo;
                int ti = sinds[selrow][q];
                sinds[selrow][q] = sinds[selrow][q - 1];
                sinds[selrow][q - 1] = ti;
              }
            }
            vmin = svals[selrow][KNN - 1];
          }
        }
      }
    }
    __syncthreads();
  }
  if (sel) {
    int* op = idxo + ((size_t)b * NPTS + r0 + selrow) * KNN;
#pragma unroll
    for (int q = 0; q < KNN; ++q) op[q] = sinds[selrow][q];
  }
}

// ---- build Wcat (2O x Cp): [Wdiff ; Wctr - Wdiff], zero-padded to Cp -------
__global__ void k_wcat(const float* __restrict__ W, int O, int Cs, int Cp,
                       float* __restrict__ wcat) {
  int t = blockIdx.x * blockDim.x + threadIdx.x;
  if (t >= 2 * O * Cp) return;
  int c = t % Cp, o2 = t / Cp;
  float v = 0.f;
  if (c < Cs) {
    if (o2 < O) v = W[(size_t)o2 * 2 * Cs + c];
    else {
      int o = o2 - O;
      v = W[(size_t)o * 2 * Cs + Cs + c] - W[(size_t)o * 2 * Cs + c];
    }
  }
  wcat[t] = v;
}

// ---- pipelined WMMA GEMM: out(M x Nout) = X(M x K, ld=ldx) * Wt(Nout x K)^T
// grid: (M/16, Nout/128), block 256; one 16x16 tile per wave; the 16 X rows
// are staged cooperatively into double-buffered LDS (shared by all 8 waves),
// B (weights) double-buffered in registers; 8 WMMA per K-chunk of 32.
// mode 0: plain store. mode 1: affine+lrelu+encoded atomicMax pool (xp).
template <int K>
__global__ __launch_bounds__(256, 1) void k_gemm_tpl(
    const float* __restrict__ X, int ldx,
    const float* __restrict__ Wt, int Nout,
    float* __restrict__ out,
    const float* __restrict__ g,
    const float* __restrict__ bb,
    unsigned* __restrict__ xp, int mode) {
  int tid = threadIdx.x;
  int wave = tid >> 5, lane = tid & 31;
  int half = lane >> 4, l15 = lane & 15;
  size_t row = (size_t)blockIdx.x * 16;
  int jt = (blockIdx.y * 8 + wave) * 16;
  const float* brow = Wt + (size_t)(jt + l15) * K + half * 2;
  v8f acc = {};

  if constexpr (K <= KC) {
    const float* arow = X + (row + l15) * (size_t)ldx + half * 2;
    v2f a[K / 4], b[K / 4];
#pragma unroll
    for (int s = 0; s < K / 4; ++s) {
      a[s] = *(const v2f*)(arow + 4 * s);
      b[s] = *(const v2f*)(brow + 4 * s);
    }
#pragma unroll
    for (int s = 0; s < K / 4; ++s) acc = wmma_f32_16x16x4(a[s], b[s], acc);
  } else {
    __shared__ float As[2][16][KC + 2];      // +2 pad: conflict-free lanes
    constexpr int NCH = K / KC;
    int ar = tid >> 4;                       // staging row 0..15
    int ac = (tid & 15) * 2;                 // staging col pair 0..30
    const float* aptr = X + (row + ar) * (size_t)ldx + ac;

    *(v2f*)&As[0][ar][ac] = *(const v2f*)(aptr);          // stage chunk 0
    v2f bbuf[2][KC / 4];
#pragma unroll
    for (int s = 0; s < KC / 4; ++s)
      bbuf[0][s] = *(const v2f*)(brow + 4 * s);
    __syncthreads();

#pragma unroll
    for (int c = 0; c < NCH; ++c) {
      int cur = c & 1, nxt = cur ^ 1;
      if (c + 1 < NCH) {                     // prefetch chunk c+1
        *(v2f*)&As[nxt][ar][ac] = *(const v2f*)(aptr + (c + 1) * KC);
#pragma unroll
        for (int s = 0; s < KC / 4; ++s)
          bbuf[nxt][s] = *(const v2f*)(brow + (c + 1) * KC + 4 * s);
      }
#pragma unroll
      for (int s = 0; s < KC / 4; ++s) {
        v2f a = *(const v2f*)&As[cur][l15][4 * s + 2 * half];
        acc = wmma_f32_16x16x4(a, bbuf[cur][s], acc);
      }
      __syncthreads();
    }
  }

  if (mode == 0) {
    float* op = out + (row + 8 * (size_t)half) * Nout + jt + l15;
#pragma unroll
    for (int r = 0; r < 8; ++r) op[(size_t)r * Nout] = acc[r];
  } else {
    int col = jt + l15;
    float gc = g[col], bc = bb[col];
    unsigned* xpb = xp + (row / NPTS) * Nout + col;
#pragma unroll
    for (int r = 0; r < 8; ++r)
      atomicMax(xpb, fenc(lrelu(acc[r] * gc + bc)));
  }
}

// ---- gather neighbors, affine+lrelu, max over k=20 -------------------------
__global__ void k_gather_max(const float* __restrict__ UV,
                             const int* __restrict__ idx,
                             const float* __restrict__ g,
                             const float* __restrict__ bb,
                             float* __restrict__ out, int O) {
  size_t t = (size_t)blockIdx.x * blockDim.x + threadIdx.x;
  int o = (int)(t % O);
  size_t bn = t / O;
  size_t two = 2 * (size_t)O;
  float vv = UV[bn * two + O + o];           // V (center) part
  size_t bbase = (bn / NPTS) * (size_t)NPTS; // batch row base
  const int* ip = idx + bn * KNN;
  float gg = g[o], bo = bb[o];
  float m = -3.4e38f;
  for (int kk = 0; kk < KNN; ++kk) {
    int j = ip[kk];
    float u = UV[(bbase + j) * two + o];     // U (neighbor) part
    m = fmaxf(m, lrelu((u + vv) * gg + bo));
  }
  out[bn * 512 + o] = m;
}

__global__ void k_init_xp(unsigned* __restrict__ xp) {
  int i = blockIdx.x * blockDim.x + threadIdx.x;
  xp[i] = 0u;  // below every encoded real value
}

// ---- final tiny MLP on (8,1024): one workgroup, LDS-resident ---------------
__global__ void k_mlp(const unsigned* __restrict__ xp_enc,
                      const float* __restrict__ wl1, const float* __restrict__ g6,
                      const float* __restrict__ b6,  const float* __restrict__ wl2,
                      const float* __restrict__ bl2, const float* __restrict__ g7,
                      const float* __restrict__ b7,  const float* __restrict__ wl3,
                      const float* __restrict__ bl3, float* __restrict__ out) {
  __shared__ float xp[BATCH * 1024];
  __shared__ float h1[BATCH * 512];
  __shared__ float h2[BATCH * 256];
  int tid = threadIdx.x;
  for (int i = tid; i < BATCH * 1024; i += 256) xp[i] = fdec(xp_enc[i]);
  __syncthreads();
  for (int p = tid; p < BATCH * 512; p += 256) {
    int b = p >> 9, o = p & 511;
    const float* wr = wl1 + (size_t)o * 1024;
    const float* xr = xp + b * 1024;
    float s = 0.f;
    for (int c = 0; c < 1024; ++c) s += xr[c] * wr[c];
    h1[p] = lrelu(s * g6[o] + b6[o]);
  }
  __syncthreads();
  for (int p = tid; p < BATCH * 256; p += 256) {
    int b = p >> 8, o = p & 255;
    const float* wr = wl2 + (size_t)o * 512;
    const float* xr = h1 + b * 512;
    float s = 0.f;
    for (int c = 0; c < 512; ++c) s += xr[c] * wr[c];
    h2[p] = lrelu((s + bl2[o]) * g7[o] + b7[o]);
  }
  __syncthreads();
  if (tid < BATCH * 7) {
    int b = tid / 7, o = tid % 7;
    const float* wr = wl3 + (size_t)o * 256;
    const float* xr = h2 + b * 256;
    float s = 0.f;
    for (int c = 0; c < 256; ++c) s += xr[c] * wr[c];
    out[tid] = s + bl3[o];
  }
}

// ---- per-layer driver (Cp = padded input channels, compile-time) -----------
template <int CP>
static void run_layer(const float* X, int ldx, int Cs, int O,
                      const float* W, const float* g, const float* bb,
                      float* outBase, float* xx, int* nidx, float* wcat,
                      float* uv, hipStream_t stream) {
  const int BN = BATCH * NPTS;
  k_xx<<<BN / 256, 256, 0, stream>>>(X, ldx, CP, xx);
  dim3 kg(NPTS / 16, BATCH);
  k_knn_tpl<CP><<<kg, 256, 0, stream>>>(X, ldx, xx, nidx);
  int wtot = 2 * O * CP;
  k_wcat<<<(wtot + 255) / 256, 256, 0, stream>>>(W, O, Cs, CP, wcat);
  dim3 gg(BN / 16, (2 * O) / 128);
  k_gemm_tpl<CP><<<gg, 256, 0, stream>>>(X, ldx, wcat, 2 * O, uv,
                                         nullptr, nullptr, nullptr, 0);
  size_t gt = (size_t)BN * O;
  k_gather_max<<<(unsigned)(gt / 256), 256, 0, stream>>>(uv, nidx, g, bb,
                                                         outBase, O);
}

extern "C" void kernel_launch(void* const* d_in, const int* in_sizes, int n_in,
                              void* d_out, int out_size, void* d_ws, size_t ws_size,
                              hipStream_t stream) {
  const float* x   = (const float*)d_in[0];
  const float* W1  = (const float*)d_in[1];
  const float* g1  = (const float*)d_in[2];
  const float* b1  = (const float*)d_in[3];
  const float* W2  = (const float*)d_in[4];
  const float* g2  = (const float*)d_in[5];
  const float* b2  = (const float*)d_in[6];
  const float* W3  = (const float*)d_in[7];
  const float* g3  = (const float*)d_in[8];
  const float* b3  = (const float*)d_in[9];
  const float* W4  = (const float*)d_in[10];
  const float* g4  = (const float*)d_in[11];
  const float* b4  = (const float*)d_in[12];
  const float* W5  = (const float*)d_in[13];
  const float* g5  = (const float*)d_in[14];
  const float* b5  = (const float*)d_in[15];
  const float* wl1 = (const float*)d_in[16];
  const float* g6  = (const float*)d_in[17];
  const float* b6  = (const float*)d_in[18];
  const float* wl2 = (const float*)d_in[19];
  const float* bl2 = (const float*)d_in[20];
  const float* g7  = (const float*)d_in[21];
  const float* b7  = (const float*)d_in[22];
  const float* wl3 = (const float*)d_in[23];
  const float* bl3 = (const float*)d_in[24];
  (void)in_sizes; (void)n_in; (void)out_size; (void)ws_size;

  char* p = (char*)d_ws;
  auto carve = [&](size_t bytes) -> char* {
    char* r = p; p += (bytes + 255) & ~(size_t)255; return r;
  };
  const int BN = BATCH * NPTS;
  float*    xt4  = (float*)carve((size_t)BN * 4 * sizeof(float));
  float*    xx   = (float*)carve((size_t)BN * sizeof(float));
  int*      nidx = (int*)carve((size_t)BN * KNN * sizeof(int));
  float*    wcat = (float*)carve((size_t)512 * 128 * sizeof(float));
  float*    uv   = (float*)carve((size_t)BN * 512 * sizeof(float));
  float*    xcat = (float*)carve((size_t)BN * 512 * sizeof(float));
  unsigned* xp   = (unsigned*)carve((size_t)BATCH * 1024 * sizeof(unsigned));

  k_build_xt4<<<BN / 256, 256, 0, stream>>>(x, xt4);

  run_layer<4>  (xt4,        4,   3,   64,  W1, g1, b1, xcat + 0,
                 xx, nidx, wcat, uv, stream);
  run_layer<64> (xcat + 0,   512, 64,  64,  W2, g2, b2, xcat + 64,
                 xx, nidx, wcat, uv, stream);
  run_layer<64> (xcat + 64,  512, 64,  128, W3, g3, b3, xcat + 128,
                 xx, nidx, wcat, uv, stream);
  run_layer<128>(xcat + 128, 512, 128, 256, W4, g4, b4, xcat + 256,
                 xx, nidx, wcat, uv, stream);

  k_init_xp<<<(BATCH * 1024) / 256, 256, 0, stream>>>(xp);
  dim3 g5g(BN / 16, 1024 / 128);
  k_gemm_tpl<512><<<g5g, 256, 0, stream>>>(xcat, 512, W5, 1024, nullptr,
                                           g5, b5, xp, 1);
  k_mlp<<<1, 256, 0, stream>>>(xp, wl1, g6, b6, wl2, bl2, g7, b7, wl3, bl3,
                               (float*)d_out);
}